// MixedRidge_33749853012534
// MI455X (gfx1250) — compile-verified
//
#include <hip/hip_runtime.h>
#include <math.h>

typedef __attribute__((ext_vector_type(2))) float v2f;
typedef __attribute__((ext_vector_type(8))) float v8f;

#define B_ 32
#define N_ 512
#define D_ 64
#define T_ 3
#define APAD 4          // Ainv row stride 68 floats = 272 B (16B multiple, bank stride 4)

// scalar LDS slots (only what other waves must see)
#define SC_RDEN 0
#define SC_W0   1       // ..4: softmax weights

#define LOG_P0    (-0.69314718f)   // log(0.5)
#define LOG_DELTA (-1.79175947f)   // log((1-0.5)/3)
// noise_var = 0.25 -> 0.5/nv = 2.0, -0.5/nv = -2.0 ; lam = TAU^2 = 1 -> Ainv0 = I, logdet0 = 0

__launch_bounds__(256, 1)
__global__ void mixed_ridge_kernel(const float* __restrict__ data,
                                   const float* __restrict__ targets,
                                   const float* __restrict__ Wm,
                                   float* __restrict__ preds) {
    // ~158.5 KB static LDS: whole batch resident, recurrence never touches HBM
    __shared__ __align__(16) float dataS[N_ * D_];      // 128 KB staged batch
    __shared__ __align__(16) float ysS[N_];             // targets
    __shared__ __align__(16) float xwTab[N_ * T_];      // precomputed x_i . W_t
    __shared__ __align__(16) float Ainv[D_][D_ + APAD]; // running A^-1
    __shared__ __align__(16) float xtys[D_];
    __shared__ __align__(16) float zs[D_];
    __shared__ __align__(16) float us[D_];
    __shared__ __align__(16) float zbuf[D_];            // zeros for B-frag lanes n>=2
    __shared__ __align__(16) float Ws[T_][D_];
    __shared__ float sc[8];

    const int b    = blockIdx.x;
    const int tid  = threadIdx.x;
    const int lane = tid & 31;
    const int wave = tid >> 5;

    const float* xb = data    + (size_t)b * N_ * D_;
    const float* yb = targets + (size_t)b * N_;

    // ---- one-time init: stage batch, W, state ----
    for (int idx = tid; idx < (N_ * D_) / 4; idx += 256)
        ((float4*)dataS)[idx] = ((const float4*)xb)[idx];           // coalesced b128
    for (int idx = tid; idx < N_; idx += 256) ysS[idx] = yb[idx];
    for (int idx = tid; idx < D_ * D_; idx += 256) {
        int r = idx >> 6, c = idx & 63;
        Ainv[r][c] = (r == c) ? 1.0f : 0.0f;                        // lam = 1 -> Ainv = I
    }
    for (int idx = tid; idx < D_; idx += 256) { xtys[idx] = 0.0f; zbuf[idx] = 0.0f; }
    for (int idx = tid; idx < D_ * T_; idx += 256) {
        int j = idx / T_, t = idx - j * T_;
        Ws[t][j] = Wm[j * T_ + t];                                  // W is (d, T) row-major
    }
    if (tid < 8) sc[tid] = 0.0f;
    __syncthreads();

    // ---- one-time: xwTab = data @ W (independent of the recurrence) ----
    for (int ii = tid; ii < N_; ii += 256) {
        const float4* xr = (const float4*)&dataS[ii * D_];
        const float4* w0 = (const float4*)&Ws[0][0];
        const float4* w1 = (const float4*)&Ws[1][0];
        const float4* w2 = (const float4*)&Ws[2][0];
        float a0 = 0.0f, a1 = 0.0f, a2 = 0.0f;
        #pragma unroll
        for (int q = 0; q < D_ / 4; ++q) {
            float4 xv = xr[q], wa = w0[q], wb = w1[q], wc = w2[q];
            a0 += xv.x * wa.x + xv.y * wa.y + xv.z * wa.z + xv.w * wa.w;
            a1 += xv.x * wb.x + xv.y * wb.y + xv.z * wb.z + xv.w * wb.w;
            a2 += xv.x * wc.x + xv.y * wc.y + xv.z * wc.z + xv.w * wc.w;
        }
        xwTab[ii * T_ + 0] = a0;
        xwTab[ii * T_ + 1] = a1;
        xwTab[ii * T_ + 2] = a2;
    }
    __syncthreads();

    // lane-0 private prefix state (registers, not LDS)
    float r_y2 = 0.0f, r_logdet = 0.0f, r_r0 = 0.0f, r_r1 = 0.0f, r_r2 = 0.0f;

    for (int i = 0; i < N_; ++i) {
        const float* xrow = &dataS[i * D_];

        // ---- Phase A: [z | u] = Ainv x [xty | x], fp32 WMMA, 16-row block per wave ----
        if (wave < 4) {
            const int rowBase = wave * 16;
            const int n  = lane & 15;          // M for A-frag, N for B-frag
            const int hk = (lane >> 4) << 1;   // 0 (lanes 0-15) or 2 (lanes 16-31)
            // per-lane B-column source: col0 = xty, col1 = x, cols 2..15 = 0
            const float* bsrc = (n == 0) ? (const float*)xtys
                               : (n == 1) ? xrow
                                          : (const float*)zbuf;
            v8f acc0 = {};
            v8f acc1 = {};
            #pragma unroll
            for (int kc = 0; kc < 16; ++kc) {
                const int kb = kc * 4 + hk;    // even -> 8B-aligned ds_load_b64
                v2f a  = *(const v2f*)&Ainv[rowBase + n][kb];
                v2f bb = *(const v2f*)&bsrc[kb];
                if (kc & 1)
                    acc1 = __builtin_amdgcn_wmma_f32_16x16x4_f32(
                        false, a, false, bb, (short)0, acc1, false, false);
                else
                    acc0 = __builtin_amdgcn_wmma_f32_16x16x4_f32(
                        false, a, false, bb, (short)0, acc0, false, false);
            }
            v8f acc = acc0 + acc1;
            // D layout: VGPR v: lanes0-15 -> (M=v, N=lane); lanes16-31 -> (M=8+v, N=lane-16)
            const int col = lane & 15;         // N==0 -> z, N==1 -> u
            if (col < 2) {
                float* dst = (col == 0) ? zs : us;
                const int rb = rowBase + ((lane >> 4) << 3);
                #pragma unroll
                for (int v = 0; v < 8; ++v) dst[rb + v] = acc[v];
            }
        }
        __syncthreads();

        // ---- Phase B (wave 0): quad & s reductions, softmax, scalar prefix update ----
        if (wave == 0) {
            float pq = zs[lane] * xtys[lane] + zs[lane + 32] * xtys[lane + 32];
            float ps = xrow[lane] * us[lane] + xrow[lane + 32] * us[lane + 32];
            #pragma unroll
            for (int off = 16; off > 0; off >>= 1) {
                pq += __shfl_down(pq, off, 32);
                ps += __shfl_down(ps, off, 32);
            }
            if (lane == 0) {
                float lg = 2.0f * pq - r_y2 + LOG_P0 - 0.5f * r_logdet;
                float l0 = -2.0f * r_r0 + LOG_DELTA;
                float l1 = -2.0f * r_r1 + LOG_DELTA;
                float l2 = -2.0f * r_r2 + LOG_DELTA;
                float mx = fmaxf(fmaxf(l0, l1), fmaxf(l2, lg));
                float e0 = __expf(l0 - mx), e1 = __expf(l1 - mx);
                float e2 = __expf(l2 - mx), e3 = __expf(lg - mx);
                float rs = 1.0f / (e0 + e1 + e2 + e3);
                sc[SC_W0]     = e0 * rs;
                sc[SC_W0 + 1] = e1 * rs;
                sc[SC_W0 + 2] = e2 * rs;
                sc[SC_W0 + 3] = e3 * rs;
                // fold point i into prefix state
                float y  = ysS[i];
                float d0 = y - xwTab[i * T_ + 0];
                float d1 = y - xwTab[i * T_ + 1];
                float d2 = y - xwTab[i * T_ + 2];
                r_y2 += y * y;
                r_r0 += d0 * d0;
                r_r1 += d1 * d1;
                r_r2 += d2 * d2;
                float denom = 1.0f + ps;            // Sherman-Morrison denominator
                r_logdet += __logf(denom);          // logdet += log(1 + x^T A^-1 x)
                sc[SC_RDEN] = 1.0f / denom;
            }
        }
        __syncthreads();

        // ---- Phase C: rank-1 Ainv update (all threads, b128) + pred + xty update ----
        {
            const int r  = tid >> 2;
            const int cb = (tid & 3) << 4;
            const float un = us[r] * sc[SC_RDEN];
            float4* arow     = (float4*)&Ainv[r][cb];   // 16B aligned (stride 272B)
            const float4* uv = (const float4*)&us[cb];
            #pragma unroll
            for (int q = 0; q < 4; ++q) {
                float4 av = arow[q];
                float4 u4 = uv[q];
                av.x -= un * u4.x;
                av.y -= un * u4.y;
                av.z -= un * u4.z;
                av.w -= un * u4.w;
                arow[q] = av;                           // Ainv -= u u^T / (1 + x^T u)
            }
        }
        if (wave == 0) {
            float w0 = sc[SC_W0],     w1 = sc[SC_W0 + 1];
            float w2 = sc[SC_W0 + 2], w3 = sc[SC_W0 + 3];
            float p = 0.0f;
            #pragma unroll
            for (int h = 0; h < 2; ++h) {
                int j = lane + h * 32;
                float wsj = Ws[0][j] * w0 + Ws[1][j] * w1 + Ws[2][j] * w2 + zs[j] * w3;
                p += xrow[j] * wsj;
            }
            #pragma unroll
            for (int off = 16; off > 0; off >>= 1) p += __shfl_down(p, off, 32);
            if (lane == 0) preds[(size_t)b * N_ + i] = p;
        } else if (wave == 1) {
            float y = ysS[i];
            xtys[lane]      += xrow[lane] * y;
            xtys[lane + 32] += xrow[lane + 32] * y;
        }
        __syncthreads();
    }
}

extern "C" void kernel_launch(void* const* d_in, const int* in_sizes, int n_in,
                              void* d_out, int out_size, void* d_ws, size_t ws_size,
                              hipStream_t stream) {
    const float* data    = (const float*)d_in[0];
    const float* targets = (const float*)d_in[1];
    const float* W       = (const float*)d_in[2];
    float* preds = (float*)d_out;
    (void)in_sizes; (void)n_in; (void)out_size; (void)d_ws; (void)ws_size;
    hipLaunchKernelGGL(mixed_ridge_kernel, dim3(B_), dim3(256), 0, stream,
                       data, targets, W, preds);
}